// Sinkhorn_seq_29824252903730
// MI455X (gfx1250) — compile-verified
//
#include <hip/hip_runtime.h>

// ---------------------------------------------------------------------------
// Sinkhorn loss for MI455X (gfx1250, wave32).
//
// Pipeline (all f32, numerics faithful to the reference):
//   K1: per-row softmax stats (max, 1/sumexp) for y_s/T and y_t/T
//   K2: tiled L1 cdist -> W[b,512,512], and K = exp(-min(W,10)*10) + 1e-8
//       (128x128 output tiles, V chunked by 128 through LDS; async-to-LDS
//        staging via global_load_async_to_lds_b128)
//   K3: u = v = 1
//   10x { rowpass: u_i /= max(u_i * (K v)_i, 1e-8)    [WMMA f32 16x16x4]
//         colpass: v_j /= max(v_j * (K^T u)_j, 1e-8)  [WMMA f32 16x16x4] }
//       (P = diag(u) K diag(v) is never materialized)
//   K6/K7: loss = (0.001/16) * sum_b sum_ij u_i K_ij v_j W_ij (deterministic
//          two-stage tree reduction)
// ---------------------------------------------------------------------------

#define Bn 16
#define Sn 512
#define Vn 4096
#define TINY 1e-8f

typedef float v2f __attribute__((ext_vector_type(2)));
typedef float v8f __attribute__((ext_vector_type(8)));
typedef int   v4i __attribute__((ext_vector_type(4)));

// ---- workspace layout (float offsets) -------------------------------------
#define WS_STATS 0u                    // 2*Bn*Sn rows * 2 = 32768 floats
#define WS_U     32768u                // Bn*Sn = 8192
#define WS_V     40960u                // Bn*Sn = 8192
#define WS_PART  49152u                // 128 partials
#define WS_W     65536u                // Bn*Sn*Sn = 4194304
#define WS_K     (65536u + 4194304u)   // Bn*Sn*Sn = 4194304
// total ~33.8 MB

// ---- gfx1250 async global->LDS path (guarded; fallback is synchronous) ----
#if __has_builtin(__builtin_amdgcn_global_load_async_to_lds_b128) && \
    __has_builtin(__builtin_amdgcn_s_wait_asynccnt)
#define USE_ASYNC_LDS 1
#endif
// Typed AS casts via integers (clang accepts int->AS-pointer casts):
// AS1 value == generic value; generic LDS address's low 32 bits are the LDS
// offset, which is exactly the AS3 representation.
#define TO_AS1(p) ((__attribute__((address_space(1))) v4i*)(unsigned long long)(p))
#define TO_AS3(p) ((__attribute__((address_space(3))) v4i*)(unsigned int)(unsigned long long)(p))

// ===========================================================================
// K1: softmax stats. One wave per row; rows [0,8192) = y_s, [8192,16384) = y_t
// ===========================================================================
__global__ __launch_bounds__(256) void softmax_stats_kernel(
    const float* __restrict__ ys, const float* __restrict__ yt,
    float* __restrict__ stats)
{
    const int t    = threadIdx.x;
    const int wave = t >> 5, lane = t & 31;
    const int row  = blockIdx.x * 8 + wave;           // 0..16383
    const float* src = (row < Bn * Sn) ? (ys + (size_t)row * Vn)
                                       : (yt + (size_t)(row - Bn * Sn) * Vn);
    float m = -3.402823466e38f;
    for (int c = lane; c < Vn; c += 32) m = fmaxf(m, src[c] * 0.5f);
    for (int o = 16; o; o >>= 1) m = fmaxf(m, __shfl_xor(m, o, 32));
    float s = 0.f;
    for (int c = lane; c < Vn; c += 32) s += __expf(src[c] * 0.5f - m);
    for (int o = 16; o; o >>= 1) s += __shfl_xor(s, o, 32);
    if (lane == 0) { stats[row * 2] = m; stats[row * 2 + 1] = 1.0f / s; }
}

// ===========================================================================
// K2: cdist + K.  grid (jblk=4, iblk=4, b=16), 256 threads, 128x128 tile,
// V chunked by 128 through LDS (row stride padded to 132 for bank spread).
// ===========================================================================
#define LDST 132
#define VC   128

__device__ __forceinline__ void load_chunk(
    float* __restrict__ lds, const float* __restrict__ src,
    const float* __restrict__ stats, int srow0, size_t gbase, int vc, int t)
{
#ifdef USE_ASYNC_LDS
    for (int idx = t; idx < 128 * 32; idx += 256) {
        const int r = idx >> 5, q = (idx & 31) << 2;
        const float* g = src + gbase + (size_t)r * Vn + vc + q;
        if (vc + VC < Vn) __builtin_prefetch(g + VC, 0, 1);   // global_prefetch_b8
        __builtin_amdgcn_global_load_async_to_lds_b128(TO_AS1(g),
                                                       TO_AS3(&lds[r * LDST + q]), 0, 0);
    }
    __builtin_amdgcn_s_wait_asynccnt(0);
    // transform own elements in place (same idx mapping -> no barrier needed)
    for (int idx = t; idx < 128 * 32; idx += 256) {
        const int r = idx >> 5, q = (idx & 31) << 2;
        float4 d = *(float4*)&lds[r * LDST + q];
        const float m  = stats[(srow0 + r) * 2];
        const float is = stats[(srow0 + r) * 2 + 1];
        d.x = __expf(d.x * 0.5f - m) * is;
        d.y = __expf(d.y * 0.5f - m) * is;
        d.z = __expf(d.z * 0.5f - m) * is;
        d.w = __expf(d.w * 0.5f - m) * is;
        *(float4*)&lds[r * LDST + q] = d;
    }
#else
    for (int idx = t; idx < 128 * 32; idx += 256) {
        const int r = idx >> 5, q = (idx & 31) << 2;
        const float* g = src + gbase + (size_t)r * Vn + vc + q;
        if (vc + VC < Vn) __builtin_prefetch(g + VC, 0, 1);
        float4 d = *(const float4*)g;
        const float m  = stats[(srow0 + r) * 2];
        const float is = stats[(srow0 + r) * 2 + 1];
        d.x = __expf(d.x * 0.5f - m) * is;
        d.y = __expf(d.y * 0.5f - m) * is;
        d.z = __expf(d.z * 0.5f - m) * is;
        d.w = __expf(d.w * 0.5f - m) * is;
        *(float4*)&lds[r * LDST + q] = d;
    }
#endif
}

__global__ __launch_bounds__(256) void cdist_kernel(
    const float* __restrict__ ys, const float* __restrict__ yt,
    const float* __restrict__ stats,
    float* __restrict__ W, float* __restrict__ Km)
{
    __shared__ float xs[128 * LDST];
    __shared__ float yl[128 * LDST];

    const int t    = threadIdx.x;
    const int tx   = t & 15;        // j lane group
    const int ty   = t >> 4;        // i lane group
    const int jblk = blockIdx.x, iblk = blockIdx.y, b = blockIdx.z;

    const int xrow0 = b * Sn + iblk * 128;            // global x rows (stats idx too)
    const int yrow0 = b * Sn + jblk * 128;
    const size_t xg  = (size_t)xrow0 * Vn;
    const size_t ygb = (size_t)yrow0 * Vn;

    float acc[8][8];
#pragma unroll
    for (int a = 0; a < 8; ++a)
#pragma unroll
        for (int q = 0; q < 8; ++q) acc[a][q] = 0.f;

    for (int vc = 0; vc < Vn; vc += VC) {
        load_chunk(xs, ys, stats, xrow0, xg, vc, t);
        load_chunk(yl, yt, stats, Bn * Sn + yrow0, ygb, vc, t);
        __syncthreads();
#pragma unroll 4
        for (int v4 = 0; v4 < VC; v4 += 4) {
            float4 xv[8], yv[8];
#pragma unroll
            for (int a = 0; a < 8; ++a) xv[a] = *(const float4*)&xs[(ty + 16 * a) * LDST + v4];
#pragma unroll
            for (int q = 0; q < 8; ++q) yv[q] = *(const float4*)&yl[(tx + 16 * q) * LDST + v4];
#pragma unroll
            for (int a = 0; a < 8; ++a)
#pragma unroll
                for (int q = 0; q < 8; ++q) {
                    acc[a][q] += fabsf(xv[a].x - yv[q].x) + fabsf(xv[a].y - yv[q].y)
                               + fabsf(xv[a].z - yv[q].z) + fabsf(xv[a].w - yv[q].w);
                }
        }
        __syncthreads();
    }

#pragma unroll
    for (int a = 0; a < 8; ++a) {
        const int i = iblk * 128 + ty + 16 * a;
#pragma unroll
        for (int q = 0; q < 8; ++q) {
            const int j = jblk * 128 + tx + 16 * q;
            const size_t o = ((size_t)b * Sn + i) * Sn + j;
            const float s = acc[a][q];
            W[o]  = s;
            Km[o] = __expf(-fminf(s, 10.f) * 10.f) + TINY;   // = exp(-min(W,10)/0.1)+1e-8
        }
    }
}

// ===========================================================================
// K3: u = v = 1
// ===========================================================================
__global__ __launch_bounds__(256) void set_ones_kernel(float* __restrict__ p, int n)
{
    const int i = blockIdx.x * 256 + threadIdx.x;
    if (i < n) p[i] = 1.0f;
}

// ===========================================================================
// Sinkhorn passes with V_WMMA_F32_16X16X4_F32 (full f32 precision).
// Layouts (ISA 7.12.2): A 16x4: VGPR0 = k{0|2}, VGPR1 = k{1|3} (lanes 0-15 |
// 16-31).  B 4x16: VGPR0 = rows {0|2}, VGPR1 = rows {1|3}, N = lane&15.
// C/D 16x16: VGPR r = rows {r | r+8}, N = lane&15.
// Broadcast operands are CONTIGUOUS pairs in LDS: {j0,j0+1} (lo half) or
// {j0+2,j0+3} (hi half) -> single aligned ds_load_b64, no register repacking
// between chained WMMAs.
// rowpass: D = A(K tile) x B(v broadcast)  -> every column of D = (K v) rows
// colpass: D = A(u broadcast) x B(K tile)  -> every row of D = (K^T u) cols
// ===========================================================================
__global__ __launch_bounds__(256) void sinkhorn_row_kernel(
    const float* __restrict__ Km, float* __restrict__ u, const float* __restrict__ v)
{
    __shared__ float vl[Sn];
    const int t = threadIdx.x, b = blockIdx.y;
    for (int c = t; c < Sn; c += 256) vl[c] = v[b * Sn + c];
    __syncthreads();

    const int wave = t >> 5, lane = t & 31;
    const int m15 = lane & 15, hi = lane >> 4;
    const int i0 = blockIdx.x * 128 + wave * 16;
    const float* Kb = Km + (size_t)b * Sn * Sn;

    v8f c = {0.f, 0.f, 0.f, 0.f, 0.f, 0.f, 0.f, 0.f};
    for (int j0 = 0; j0 < Sn; j0 += 4) {
        // A: lane m15 holds row i0+m15, cols (j0, j0+1) or (j0+2, j0+3)
        v2f a = *(const v2f*)(Kb + (size_t)(i0 + m15) * Sn + j0 + 2 * hi);
        // B[k][n] = v[j0+k] for all n: contiguous pair -> one ds_load_b64
        v2f bb = *(const v2f*)&vl[j0 + 2 * hi];
        c = __builtin_amdgcn_wmma_f32_16x16x4_f32(false, a, false, bb,
                                                  (short)0, c, false, false);
    }
    // lane 0 holds rows i0..i0+7 in c[0..7]; lane 16 holds rows i0+8..i0+15
    if (lane == 0 || lane == 16) {
        const int base = b * Sn + i0 + hi * 8;
#pragma unroll
        for (int m = 0; m < 8; ++m) {
            const float uo = u[base + m];
            const float r  = fmaxf(uo * c[m], TINY);
            u[base + m] = uo / r;
        }
    }
}

__global__ __launch_bounds__(256) void sinkhorn_col_kernel(
    const float* __restrict__ Km, const float* __restrict__ u, float* __restrict__ v)
{
    __shared__ float ul[Sn];
    const int t = threadIdx.x, b = blockIdx.y;
    for (int c = t; c < Sn; c += 256) ul[c] = u[b * Sn + c];
    __syncthreads();

    const int wave = t >> 5, lane = t & 31;
    const int m15 = lane & 15, hi = lane >> 4;
    const int j0 = blockIdx.x * 128 + wave * 16;
    const float* Kb = Km + (size_t)b * Sn * Sn;

    v8f c = {0.f, 0.f, 0.f, 0.f, 0.f, 0.f, 0.f, 0.f};
    for (int k0 = 0; k0 < Sn; k0 += 4) {
        const int r0 = k0 + 2 * hi;
        // B tile: natural row-major rows of K (coalesced across lanes 0-15)
        v2f bb;
        bb.x = Kb[(size_t)r0 * Sn + j0 + m15];
        bb.y = Kb[(size_t)(r0 + 1) * Sn + j0 + m15];
        // A[m][k] = u[k0+k] for all m: contiguous pair -> one ds_load_b64
        v2f a = *(const v2f*)&ul[r0];
        c = __builtin_amdgcn_wmma_f32_16x16x4_f32(false, a, false, bb,
                                                  (short)0, c, false, false);
    }
    // D row 0: c[0] on lanes 0-15 holds column j0+lane
    if (lane < 16) {
        const int j = b * Sn + j0 + lane;
        const float vo = v[j];
        const float r  = fmaxf(vo * c[0], TINY);
        v[j] = vo / r;
    }
}

// ===========================================================================
// K6: per-(b, 64-row block) partial of sum u_i K_ij v_j W_ij
// ===========================================================================
__global__ __launch_bounds__(256) void loss_partial_kernel(
    const float* __restrict__ Km, const float* __restrict__ W,
    const float* __restrict__ u, const float* __restrict__ v,
    float* __restrict__ partials)
{
    const int t = threadIdx.x, b = blockIdx.y;
    const int i0 = blockIdx.x * 64;
    float p = 0.f;
    for (int idx = t; idx < 64 * Sn; idx += 256) {
        const int i = i0 + (idx >> 9);
        const int j = idx & (Sn - 1);
        const size_t o = ((size_t)b * Sn + i) * Sn + j;
        p += u[b * Sn + i] * Km[o] * v[b * Sn + j] * W[o];
    }
    for (int o = 16; o; o >>= 1) p += __shfl_xor(p, o, 32);
    __shared__ float red[8];
    if ((t & 31) == 0) red[t >> 5] = p;
    __syncthreads();
    if (t == 0) {
        float s = 0.f;
#pragma unroll
        for (int w = 0; w < 8; ++w) s += red[w];
        partials[b * 8 + blockIdx.x] = s;
    }
}

// K7: deterministic final reduction of 128 partials -> mean(0.001 * loss_b)
__global__ __launch_bounds__(128) void final_reduce_kernel(
    const float* __restrict__ partials, float* __restrict__ out)
{
    const int t = threadIdx.x;
    float p = partials[t];
    for (int o = 16; o; o >>= 1) p += __shfl_xor(p, o, 32);
    __shared__ float red[4];
    if ((t & 31) == 0) red[t >> 5] = p;
    __syncthreads();
    if (t == 0) out[0] = (red[0] + red[1] + red[2] + red[3]) * (0.001f / (float)Bn);
}

// ===========================================================================
extern "C" void kernel_launch(void* const* d_in, const int* in_sizes, int n_in,
                              void* d_out, int out_size, void* d_ws, size_t ws_size,
                              hipStream_t stream)
{
    (void)in_sizes; (void)n_in; (void)out_size; (void)ws_size;
    const float* ys = (const float*)d_in[0];
    const float* yt = (const float*)d_in[1];
    float* ws  = (float*)d_ws;
    float* out = (float*)d_out;

    float* stats = ws + WS_STATS;
    float* u     = ws + WS_U;
    float* v     = ws + WS_V;
    float* part  = ws + WS_PART;
    float* W     = ws + WS_W;
    float* Km    = ws + WS_K;

    softmax_stats_kernel<<<2 * Bn * Sn / 8, 256, 0, stream>>>(ys, yt, stats);
    cdist_kernel<<<dim3(4, 4, Bn), 256, 0, stream>>>(ys, yt, stats, W, Km);
    set_ones_kernel<<<(2 * Bn * Sn + 255) / 256, 256, 0, stream>>>(u, 2 * Bn * Sn);

    for (int it = 0; it < 10; ++it) {
        sinkhorn_row_kernel<<<dim3(4, Bn), 256, 0, stream>>>(Km, u, v);
        sinkhorn_col_kernel<<<dim3(4, Bn), 256, 0, stream>>>(Km, u, v);
    }

    loss_partial_kernel<<<dim3(8, Bn), 256, 0, stream>>>(Km, W, u, v, part);
    final_reduce_kernel<<<1, 128, 0, stream>>>(part, out);
}